// GCN_1580547969573
// MI455X (gfx1250) — compile-verified
//
#include <hip/hip_runtime.h>
#include <hip/hip_bf16.h>

// ---------------- threefry2x32 (matches JAX PRNG) ----------------
#define TF_ROT(x, r) (((x) << (r)) | ((x) >> (32 - (r))))
#define TF_RA(i) do { x0 += x1; x1 = TF_ROT(x1, (i)); x1 ^= x0; } while (0)

__host__ __device__ inline void threefry2x32(unsigned k0, unsigned k1,
                                             unsigned x0, unsigned x1,
                                             unsigned* o0, unsigned* o1) {
  unsigned ks0 = k0, ks1 = k1, ks2 = 0x1BD11BDAu ^ k0 ^ k1;
  x0 += ks0; x1 += ks1;
  TF_RA(13); TF_RA(15); TF_RA(26); TF_RA(6);
  x0 += ks1; x1 += ks2 + 1u;
  TF_RA(17); TF_RA(29); TF_RA(16); TF_RA(24);
  x0 += ks2; x1 += ks0 + 2u;
  TF_RA(13); TF_RA(15); TF_RA(26); TF_RA(6);
  x0 += ks0; x1 += ks1 + 3u;
  TF_RA(17); TF_RA(29); TF_RA(16); TF_RA(24);
  x0 += ks1; x1 += ks2 + 4u;
  TF_RA(13); TF_RA(15); TF_RA(26); TF_RA(6);
  x0 += ks2; x1 += ks0 + 5u;
  *o0 = x0; *o1 = x1;
}

__device__ inline float tf_u01(unsigned bits) {
  // JAX float32 uniform: (bits >> 9) | 0x3f800000 -> [1,2), minus 1.0
  unsigned f = (bits >> 9) | 0x3f800000u;
  return __uint_as_float(f) - 1.0f;
}

// ---------------- elementwise kernels ----------------
__global__ void zero_kernel(float* __restrict__ p, int n) {
  int i = blockIdx.x * blockDim.x + threadIdx.x;
  if (i < n) p[i] = 0.0f;
}

// dropout(p=0.5): thread j handles flat elements j and j+H (one threefry per pair)
__global__ void dropout_kernel(const float* __restrict__ in, float* __restrict__ out,
                               int H, unsigned k0, unsigned k1) {
  int j = blockIdx.x * blockDim.x + threadIdx.x;
  if (j >= H) return;
  unsigned o0, o1;
  threefry2x32(k0, k1, (unsigned)j, (unsigned)(j + H), &o0, &o1);
  float a = in[j], b = in[j + H];
  out[j]     = (tf_u01(o0) < 0.5f) ? a * 2.0f : 0.0f;
  out[j + H] = (tf_u01(o1) < 0.5f) ? b * 2.0f : 0.0f;
}

// ---------------- COO SpMM: out[r,:] += w[e] * in[c,:] ----------------
// one wave per edge; lane covers 4 of 128 features via float4 + 4x atomic f32 add
__global__ void spmm_atomic(const int* __restrict__ rows, const int* __restrict__ cols,
                            const float* __restrict__ w, const float* __restrict__ hin,
                            float* __restrict__ hout, int E) {
  int e = blockIdx.x * (blockDim.x >> 5) + (threadIdx.x >> 5);
  if (e >= E) return;
  int lane = threadIdx.x & 31;
  int r = rows[e];
  int c = cols[e];
  float we = w[e];
  float4 v = *((const float4*)(hin + (size_t)c * 128) + lane);
  float* dst = hout + (size_t)r * 128 + (size_t)lane * 4;
  atomicAdd(dst + 0, v.x * we);
  atomicAdd(dst + 1, v.y * we);
  atomicAdd(dst + 2, v.z * we);
  atomicAdd(dst + 3, v.w * we);
}

// ---------------- dense layer via V_WMMA_F32_16X16X4_F32 ----------------
typedef float v2f __attribute__((ext_vector_type(2)));
typedef float v8f __attribute__((ext_vector_type(8)));

// out[N x COLS] = act( A[N x 128] @ W[128 x COLS] + bias ), optional fused dropout
// block = 32 * (COLS/16) threads; wave `w` computes cols [16w, 16w+16) of a 16-row tile
template <int COLS, bool RELU, bool DROP>
__global__ void gemm_wmma(const float* __restrict__ A, const float* __restrict__ W,
                          const float* __restrict__ bias, float* __restrict__ out,
                          int n, unsigned dk0, unsigned dk1) {
  const int lane = threadIdx.x & 31;
  const int wave = threadIdx.x >> 5;
  const int row0 = blockIdx.x * 16;
  const int c0 = wave * 16;
  const int m = lane & 15;              // M (for A) / N (for B,C,D) sub-index
  const int half = lane >> 4;           // 0: lanes 0-15, 1: lanes 16-31
  const int koff = half * 2;            // A/B K offset per ISA layout

  int arow = row0 + m;
  if (arow >= n) arow = n - 1;          // clamp (grid is exact for N=50000)
  const float* aptr = A + (size_t)arow * 128;

  v8f acc = {};
#pragma unroll
  for (int k = 0; k < 128; k += 4) {
    // A 16x4: lane half 0 holds K=k,k+1; half 1 holds K=k+2,k+3  (8B contiguous)
    v2f a = *(const v2f*)(aptr + k + koff);
    // B 4x16: VGPR0 = row (k+koff), VGPR1 = row (k+koff+1), col striped on lanes
    v2f b;
    b.x = W[(size_t)(k + koff) * COLS + c0 + m];
    b.y = W[(size_t)(k + koff + 1) * COLS + c0 + m];
    acc = __builtin_amdgcn_wmma_f32_16x16x4_f32(
        /*neg_a=*/false, a, /*neg_b=*/false, b,
        /*c_mod=*/(short)0, acc, /*reuse_a=*/false, /*reuse_b=*/false);
  }

  const int col = c0 + m;
  const float bv = bias[col];
  const int H = (n * COLS) / 2;
#pragma unroll
  for (int v = 0; v < 8; v++) {
    int row = row0 + v + half * 8;      // C/D layout: VGPR v -> M = v + 8*half
    if (row >= n) continue;
    float val = acc[v] + bv;
    if (RELU) val = fmaxf(val, 0.0f);
    if (DROP) {
      int j = row * COLS + col;
      unsigned o0, o1;
      if (j < H) {
        threefry2x32(dk0, dk1, (unsigned)j, (unsigned)(j + H), &o0, &o1);
        val = (tf_u01(o0) < 0.5f) ? val * 2.0f : 0.0f;
      } else {
        threefry2x32(dk0, dk1, (unsigned)(j - H), (unsigned)j, &o0, &o1);
        val = (tf_u01(o1) < 0.5f) ? val * 2.0f : 0.0f;
      }
    }
    out[(size_t)row * COLS + col] = val;
  }
}

// ---------------- launcher ----------------
extern "C" void kernel_launch(void* const* d_in, const int* in_sizes, int n_in,
                              void* d_out, int out_size, void* d_ws, size_t ws_size,
                              hipStream_t stream) {
  const int*   rows = (const int*)d_in[0];
  const int*   cols = (const int*)d_in[1];
  const float* w    = (const float*)d_in[2];
  const float* X    = (const float*)d_in[3];
  const float* W1   = (const float*)d_in[4];
  const float* b1   = (const float*)d_in[5];
  const float* W2   = (const float*)d_in[6];
  const float* b2   = (const float*)d_in[7];

  const int E  = in_sizes[0];
  const int NF = in_sizes[3];      // N * 128
  const int n  = NF / 128;

  float* h0 = (float*)d_ws;        // dropout(X)
  float* h1 = h0 + NF;             // A @ h0
  float* h2 = h1 + NF;             // dropout(relu(h1 W1 + b1))
  float* h3 = h2 + NF;             // A @ h2

  // host-side replication of k1, k2 = jax.random.split(jax.random.key(42))
  // key(42) = (0, 42); split -> threefry over counts [0,1 | 2,3]
  unsigned o0a, o1a, o0b, o1b;
  threefry2x32(0u, 42u, 0u, 2u, &o0a, &o1a);
  threefry2x32(0u, 42u, 1u, 3u, &o0b, &o1b);
  const unsigned k1_0 = o0a, k1_1 = o0b;   // k1 = out[0], out[1]
  const unsigned k2_0 = o1a, k2_1 = o1b;   // k2 = out[2], out[3]

  const int H = NF / 2;
  dropout_kernel<<<(H + 255) / 256, 256, 0, stream>>>(X, h0, H, k1_0, k1_1);

  zero_kernel<<<(NF + 255) / 256, 256, 0, stream>>>(h1, NF);
  spmm_atomic<<<(E + 7) / 8, 256, 0, stream>>>(rows, cols, w, h0, h1, E);

  gemm_wmma<128, true, true><<<(n + 15) / 16, 256, 0, stream>>>(
      h1, W1, b1, h2, n, k2_0, k2_1);

  zero_kernel<<<(NF + 255) / 256, 256, 0, stream>>>(h3, NF);
  spmm_atomic<<<(E + 7) / 8, 256, 0, stream>>>(rows, cols, w, h2, h3, E);

  gemm_wmma<64, false, false><<<(n + 15) / 16, 128, 0, stream>>>(
      h3, W2, b2, (float*)d_out, n, 0u, 0u);
}